// SpikingNN_78082505441725
// MI455X (gfx1250) — compile-verified
//
#include <hip/hip_runtime.h>
#include <hip/hip_bf16.h>
#include <stdint.h>

// ---------------------------------------------------------------------------
// LIF spiking-NN scan, MI455X (gfx1250, wave32).
// Memory-bound (420 MB moved, ~18 us floor @ 23.3 TB/s). Strategy: stage
// 256-trace x 32-timestep tiles through LDS using CDNA5 async global<->LDS
// copies (ASYNCcnt-tracked), compute the sequential recurrence out of LDS
// (bank-conflict-free padded stride), write back with async stores.
// ---------------------------------------------------------------------------

#define LLEN   401           // timesteps per trace
#define TS     32            // tile width (timesteps)
#define ROWS   256           // traces per workgroup ( == blockDim.x )
#define LDSS   (TS + 1)      // padded LDS row stride (33 coprime w/ 64 banks)
#define NTILES ((LLEN + TS - 1) / TS)   // 13

typedef __attribute__((address_space(1))) int* g_i32p;   // global int*
typedef __attribute__((address_space(3))) int* l_i32p;   // LDS int*

__device__ __forceinline__ void async_load_b32(const float* g, const float* l) {
    // copies 4 bytes from the global address g to the LDS address l (tracked on ASYNCcnt)
    __builtin_amdgcn_global_load_async_to_lds_b32(
        (g_i32p)(uintptr_t)g,
        (l_i32p)(uint32_t)(uintptr_t)l,   // the low 32 bits of the flat LDS pointer are the LDS offset
        0 /*imm offset*/, 0 /*cpol*/);
}

__device__ __forceinline__ void async_store_b32(float* g, const float* l) {
    // copies 4 bytes from the LDS address l to the global address g (tracked on ASYNCcnt)
    __builtin_amdgcn_global_store_async_from_lds_b32(
        (g_i32p)(uintptr_t)g,
        (l_i32p)(uint32_t)(uintptr_t)l,
        0 /*imm offset*/, 0 /*cpol*/);
}

__device__ __forceinline__ void wait_async0() {
    asm volatile("s_wait_asynccnt 0" ::: "memory");
}

__global__ __launch_bounds__(ROWS) void spiking_lif_scan(
    const float* __restrict__ S,   // [N, 401]
    float* __restrict__ out,       // [N, 401]
    int N)
{
    __shared__ float lin[ROWS * LDSS];
    __shared__ float lout[ROWS * LDSS];

    const int tid  = threadIdx.x;            // 0..255
    const int r0   = blockIdx.x * ROWS;

    // Coalesced copy mapping: lanes 0..31 of a wave cover 32 consecutive
    // columns of one row (128 contiguous bytes); 8 rows per 256-thread pass.
    const int colL = tid & (TS - 1);         // 0..31
    const int rowB = tid >> 5;               // 0..7

    // Per-trace recurrence state (thread tid owns trace r0 + tid)
    float P = 0.0f;
    float t_rest = 0.0f;

    for (int k = 0; k < NTILES; ++k) {
        const int t0   = k * TS;
        const int tcol = t0 + colL;

        // -------- issue async loads for tile k (EXEC-masked guards) --------
        if (tcol < LLEN) {
            #pragma unroll
            for (int pass = 0; pass < ROWS / 8; ++pass) {
                const int rl = pass * 8 + rowB;
                const int r  = r0 + rl;
                if (r < N) {
                    async_load_b32(S + (size_t)r * LLEN + tcol,
                                   &lin[rl * LDSS + colL]);
                }
            }
        }

        // Wait own wave's async ops (this tile's loads + previous tile's
        // stores), then barrier so every wave's transfers are retired before
        // anyone touches lin (read) or lout (overwrite).
        wait_async0();
        __syncthreads();

        // -------- compute: sequential LIF steps for this thread's trace ----
        const int tmax = (t0 + TS <= LLEN) ? TS : (LLEN - t0);
        for (int tc = 0; tc < tmax; ++tc) {
            const int   i = t0 + tc;
            const float s = lin[tid * LDSS + tc];     // stride-33: no bank conflicts
            const float t = (float)i * 0.125f;        // TIME[i], exact in f32
            float Pn;
            if (i == 0) {
                Pn = s - 0.25f;                       // P0 = S[:,0] - D
            } else {
                Pn = (t <= t_rest) ? 0.0f
                     : ((P > -1.0f) ? (P + s - 0.25f) : 0.0f);
            }
            if (Pn >= 25.0f) {                        // spike
                Pn    += 4.0f;                        // overshoot marker
                t_rest = t + 5.0f;                    // refractory window
            }
            P = Pn;
            lout[tid * LDSS + tc] = Pn;
        }
        __syncthreads();

        // -------- issue async stores for tile k (coalesced writeback) ------
        if (tcol < LLEN) {
            #pragma unroll
            for (int pass = 0; pass < ROWS / 8; ++pass) {
                const int rl = pass * 8 + rowB;
                const int r  = r0 + rl;
                if (r < N) {
                    async_store_b32(out + (size_t)r * LLEN + tcol,
                                    &lout[rl * LDSS + colL]);
                }
            }
        }
        // No wait here: next iteration's wait_async0 (before lout overwrite)
        // retires these; s_endpgm's implicit wait-idle drains the final tile.
    }
}

extern "C" void kernel_launch(void* const* d_in, const int* in_sizes, int n_in,
                              void* d_out, int out_size, void* d_ws, size_t ws_size,
                              hipStream_t stream) {
    // setup_inputs order: d_in[0] = Pn (dead, never read), d_in[1] = S
    const float* S  = (const float*)d_in[1];
    float* out      = (float*)d_out;
    const int N     = in_sizes[1] / LLEN;    // 131072

    const int grid = (N + ROWS - 1) / ROWS;  // 512 workgroups
    spiking_lif_scan<<<grid, ROWS, 0, stream>>>(S, out, N);
}